// GMM_50422916055543
// MI455X (gfx1250) — compile-verified
//
#include <hip/hip_runtime.h>
#include <hip/hip_bf16.h>
#include <math.h>

typedef __attribute__((ext_vector_type(2))) float v2f;
typedef __attribute__((ext_vector_type(8))) float v8f;

#define GMM_N 524288
#define GMM_K 32
#define GMM_D 8
#define NFEAT 48          // 36 sym products + 8 linear + 1 const + 3 zero pad
#define GMM_EPS 1e-6f

// Upper-triangle (i<=j) pair enumeration for the 36 quadratic features.
constexpr int FI[36] = {0,0,0,0,0,0,0,0, 1,1,1,1,1,1,1, 2,2,2,2,2,2,
                        3,3,3,3,3, 4,4,4,4, 5,5,5, 6,6, 7};
constexpr int FJ[36] = {0,1,2,3,4,5,6,7, 1,2,3,4,5,6,7, 2,3,4,5,6,7,
                        3,4,5,6,7, 4,5,6,7, 5,6,7, 6,7, 7};

// ---------------------------------------------------------------------------
// Precompute: one wave, lane k handles component k. Builds the effective
// weight matrix W[f][k] (wll[n,k] = sum_f phi[n,f] * W[f][k]) in LDS, then
// repacks it into B-fragment order so the main kernel loads one float4 per
// lane per k-step:
//   Wp[s*32 + lane] = { W[f0][c], W[f0+1][c], W[f0][c+16], W[f0+1][c+16] }
//   with c = lane&15, f0 = 4*s + 2*(lane>>4)
// matching the 4x16 f32 B layout (VGPR0 = rows K0/K2, VGPR1 = rows K1/K3).
// ---------------------------------------------------------------------------
__global__ __launch_bounds__(32)
void gmm_precompute(const float* __restrict__ pi,
                    const float* __restrict__ means,
                    const float* __restrict__ cholv,
                    float* __restrict__ Wp) {
    __shared__ float ldsW[NFEAT * GMM_K];
    const int k = threadIdx.x;   // 0..31

    // L = tril(chol_var[k])
    float L[GMM_D][GMM_D];
    for (int i = 0; i < GMM_D; ++i)
        for (int j = 0; j < GMM_D; ++j)
            L[i][j] = (j <= i) ? cholv[k * 64 + i * 8 + j] : 0.0f;

    // cov = L L^T + eps I
    float cov[GMM_D][GMM_D];
    for (int i = 0; i < GMM_D; ++i)
        for (int j = 0; j < GMM_D; ++j) {
            float s = 0.0f;
            for (int t = 0; t < GMM_D; ++t) s += L[i][t] * L[j][t];
            cov[i][j] = s + ((i == j) ? GMM_EPS : 0.0f);
        }

    // C = cholesky(cov), lower
    float C[GMM_D][GMM_D];
    for (int i = 0; i < GMM_D; ++i)
        for (int j = 0; j < GMM_D; ++j) C[i][j] = 0.0f;
    for (int i = 0; i < GMM_D; ++i) {
        for (int j = 0; j <= i; ++j) {
            float s = cov[i][j];
            for (int t = 0; t < j; ++t) s -= C[i][t] * C[j][t];
            if (i == j) C[i][j] = sqrtf(s);
            else        C[i][j] = s / C[j][j];
        }
    }
    float logdet = 0.0f;
    for (int i = 0; i < GMM_D; ++i) logdet += logf(C[i][i]);
    logdet *= 2.0f;

    // Ci = C^-1 (lower triangular inverse)
    float Ci[GMM_D][GMM_D];
    for (int i = 0; i < GMM_D; ++i)
        for (int j = 0; j < GMM_D; ++j) Ci[i][j] = 0.0f;
    for (int j = 0; j < GMM_D; ++j) {
        Ci[j][j] = 1.0f / C[j][j];
        for (int i = j + 1; i < GMM_D; ++i) {
            float s = 0.0f;
            for (int t = j; t < i; ++t) s += C[i][t] * Ci[t][j];
            Ci[i][j] = -s / C[i][i];
        }
    }

    // A = Ci^T Ci  (precision matrix)
    float A[GMM_D][GMM_D];
    for (int i = 0; i < GMM_D; ++i)
        for (int j = 0; j < GMM_D; ++j) {
            float s = 0.0f;
            for (int t = 0; t < GMM_D; ++t) s += Ci[t][i] * Ci[t][j];
            A[i][j] = s;
        }

    // b = A m,  c = m^T A m
    float m[GMM_D], b[GMM_D];
    for (int i = 0; i < GMM_D; ++i) m[i] = means[k * 8 + i];
    float c = 0.0f;
    for (int i = 0; i < GMM_D; ++i) {
        float s = 0.0f;
        for (int j = 0; j < GMM_D; ++j) s += A[i][j] * m[j];
        b[i] = s;
        c += s * m[i];
    }

    // log_softmax(pi) across the wave (32 lanes)
    float p = pi[k];
    float mx = p;
    for (int off = 1; off < 32; off <<= 1)
        mx = fmaxf(mx, __shfl_xor(mx, off, 32));
    float se = expf(p - mx);
    for (int off = 1; off < 32; off <<= 1)
        se += __shfl_xor(se, off, 32);
    float lsm = p - mx - logf(se);

    // Column k of W into LDS
    for (int f = 0; f < 36; ++f) {
        float sc = (FI[f] == FJ[f]) ? 1.0f : 2.0f;
        ldsW[f * GMM_K + k] = -0.5f * sc * A[FI[f]][FJ[f]];
    }
    for (int i = 0; i < GMM_D; ++i)
        ldsW[(36 + i) * GMM_K + k] = b[i];
    const float LOG2PI = 1.8378770664093454836f;
    ldsW[44 * GMM_K + k] = -0.5f * c - 0.5f * (float)GMM_D * LOG2PI
                           - 0.5f * logdet + lsm;
    ldsW[45 * GMM_K + k] = 0.0f;
    ldsW[46 * GMM_K + k] = 0.0f;
    ldsW[47 * GMM_K + k] = 0.0f;

    __syncthreads();

    // Repack into B-fragment order: one float4 per (step, lane).
    const int half = k >> 4;
    const int cc   = k & 15;
    float4* Wp4 = reinterpret_cast<float4*>(Wp);
    for (int s = 0; s < 12; ++s) {
        const int f0 = s * 4 + 2 * half;
        float4 w;
        w.x = ldsW[f0 * GMM_K + cc];
        w.y = ldsW[(f0 + 1) * GMM_K + cc];
        w.z = ldsW[f0 * GMM_K + cc + 16];
        w.w = ldsW[(f0 + 1) * GMM_K + cc + 16];
        Wp4[s * 32 + k] = w;
    }
}

// ---------------------------------------------------------------------------
// Feature value for compile-time f (after full unroll all indices fold).
// ---------------------------------------------------------------------------
__device__ __forceinline__ float featval(const float xr[8], int f) {
    if (f < 36) return xr[FI[f]] * xr[FJ[f]];
    if (f < 44) return xr[f - 36];
    if (f == 44) return 1.0f;
    return 0.0f;
}

// ---------------------------------------------------------------------------
// Main: each wave computes a 16-point x 32-component tile of wll via
// V_WMMA_F32_16X16X4_F32, then a half-wave-shuffle logsumexp over K.
// ---------------------------------------------------------------------------
__global__ __launch_bounds__(256)
void gmm_main(const float* __restrict__ x,
              const float* __restrict__ Wp,
              float* __restrict__ out) {
    const int lane = threadIdx.x & 31;
    const int half = lane >> 4;                 // 0: lanes 0-15, 1: lanes 16-31
    const int wave = threadIdx.x >> 5;
    const int tile = blockIdx.x * 8 + wave;     // 16 points per tile
    const int pt   = tile * 16 + (lane & 15);   // lanes L and L+16 share a point

    // Load this lane's point (row of x), 32B aligned.
    const float4* xv = reinterpret_cast<const float4*>(x + (size_t)pt * 8);
    float4 xa = xv[0];
    float4 xb = xv[1];
    float xr[8] = {xa.x, xa.y, xa.z, xa.w, xb.x, xb.y, xb.z, xb.w};

    // B fragments: 12 coalesced b128 loads, pre-permuted by gmm_precompute.
    const float4* Wp4 = reinterpret_cast<const float4*>(Wp);
    v2f B0[12], B1[12];
#pragma unroll
    for (int s = 0; s < 12; ++s) {
        float4 w = Wp4[s * 32 + lane];
        B0[s].x = w.x; B0[s].y = w.y;   // component tile 0 (comps 0..15)
        B1[s].x = w.z; B1[s].y = w.w;   // component tile 1 (comps 16..31)
    }

    v8f acc0 = {};   // wll for comps 0..15
    v8f acc1 = {};   // wll for comps 16..31
#pragma unroll
    for (int s = 0; s < 12; ++s) {
        // A fragment (16x4 f32): VGPR0 = K0 (lanes 0-15) / K2 (lanes 16-31),
        // VGPR1 = K1 / K3. Feature index constant per branch -> folds to regs.
        v2f a;
        a.x = half ? featval(xr, s * 4 + 2) : featval(xr, s * 4 + 0);
        a.y = half ? featval(xr, s * 4 + 3) : featval(xr, s * 4 + 1);
        acc0 = __builtin_amdgcn_wmma_f32_16x16x4_f32(
            false, a, false, B0[s], (short)0, acc0, false, false);
        acc1 = __builtin_amdgcn_wmma_f32_16x16x4_f32(
            false, a, false, B1[s], (short)0, acc1, false, false);
    }

    // Epilogue: C layout -> VGPR v, lanes 0-15: point v, lanes 16-31: point v+8;
    // column N = lane&15 is the component. logsumexp over 32 comps = reduce the
    // two accumulators across each 16-lane half (xor 1,2,4,8 stays in-half).
    float rsel = 0.0f;
#pragma unroll
    for (int v = 0; v < 8; ++v) {
        float a0 = acc0[v];
        float a1 = acc1[v];
        float mx = fmaxf(a0, a1);
#pragma unroll
        for (int off = 1; off < 16; off <<= 1)
            mx = fmaxf(mx, __shfl_xor(mx, off, 32));
        float s = __expf(a0 - mx) + __expf(a1 - mx);
#pragma unroll
        for (int off = 1; off < 16; off <<= 1)
            s += __shfl_xor(s, off, 32);
        float r = mx + __logf(s);
        rsel = ((lane & 15) == v) ? r : rsel;   // lane v / v+16 keeps point v / v+8
    }
    // Lanes 0-7 hold points 0-7; lanes 16-23 hold points 8-15. Single store.
    if ((lane & 15) < 8)
        out[(long)tile * 16 + (lane & 15) + 8 * half] = rsel;
}

// ---------------------------------------------------------------------------
extern "C" void kernel_launch(void* const* d_in, const int* in_sizes, int n_in,
                              void* d_out, int out_size, void* d_ws, size_t ws_size,
                              hipStream_t stream) {
    const float* x     = (const float*)d_in[0];   // [N,8]
    const float* pi    = (const float*)d_in[1];   // [32]
    const float* means = (const float*)d_in[2];   // [32,8]
    const float* cholv = (const float*)d_in[3];   // [32,8,8]
    float* Wp  = (float*)d_ws;                    // packed B fragments, 6 KB
    float* out = (float*)d_out;                   // [N]

    gmm_precompute<<<1, 32, 0, stream>>>(pi, means, cholv, Wp);

    const int tiles  = GMM_N / 16;   // 32768 waves
    const int blocks = tiles / 8;    // 4096 blocks x 8 waves
    gmm_main<<<blocks, 256, 0, stream>>>(x, Wp, out);
}